// ProcessVGAE_43722767073853
// MI455X (gfx1250) — compile-verified
//
#include <hip/hip_runtime.h>
#include <hip/hip_bf16.h>

#define N_NODES 100000
#define N_EDGES 1600000
#define IN_F 25
#define OUT_F 50

typedef __attribute__((ext_vector_type(2))) float v2f;
typedef __attribute__((ext_vector_type(8))) float v8f;

// ---------------------------------------------------------------------------
// Kernel 1: zero the output accumulator (float4-vectorized).
// ---------------------------------------------------------------------------
__global__ void gcn_zero_kernel(float4* __restrict__ out, int n4) {
    int i = blockIdx.x * blockDim.x + threadIdx.x;
    if (i < n4) out[i] = make_float4(0.f, 0.f, 0.f, 0.f);
}

// ---------------------------------------------------------------------------
// Kernel 2: h = x @ W via V_WMMA_F32_16X16X4_F32.
// Block: 256 threads = 8 waves. Block covers 32 rows (M) x 50 cols (N).
// N_NODES = 100000 = 3125 * 32 exactly -> no M-boundary guards anywhere.
// K = 25 padded to 28 (7 k-steps of 4).
// ---------------------------------------------------------------------------
__global__ void __launch_bounds__(256)
gcn_gemm_kernel(const float* __restrict__ x, const float* __restrict__ W,
                float* __restrict__ h) {
    __shared__ float xs[32][28];   // x tile, K padded with zeros
    __shared__ float wsm[28][64];  // W, K and N padded with zeros

    const int tid = threadIdx.x;
    const int mBase = blockIdx.x * 32;

    // Zero LDS (covers the K/N padding)
    for (int i = tid; i < 32 * 28; i += 256) ((float*)xs)[i] = 0.f;
    for (int i = tid; i < 28 * 64; i += 256) ((float*)wsm)[i] = 0.f;
    __syncthreads();

    // Stage x tile: 32 rows x 25 K-values (no row guard: grid covers exactly)
    for (int i = tid; i < 32 * IN_F; i += 256) {
        int r = i / IN_F, c = i % IN_F;
        xs[r][c] = x[(mBase + r) * IN_F + c];
    }
    // Stage W: 25 x 50
    for (int i = tid; i < IN_F * OUT_F; i += 256) {
        int r = i / OUT_F, c = i % OUT_F;
        wsm[r][c] = W[i];
    }
    __syncthreads();

    const int lane  = tid & 31;
    const int wave  = tid >> 5;
    const int mSub  = wave >> 2;   // 0..1
    const int nTile = wave & 3;    // 0..3
    const int half  = lane >> 4;   // 0..1
    const int lq    = lane & 15;   // 0..15

    const int mLocal = mSub * 16 + lq;      // A-frag row for this lane
    const int nCol   = nTile * 16 + lq;     // B/C-frag column for this lane

    v8f c = {};
#pragma unroll
    for (int kk = 0; kk < 7; ++kk) {
        const int k = kk * 4 + 2 * half;
        // A 16x4 fp32 layout: lanes 0-15 hold K={k0,k0+1}, lanes 16-31 K={k0+2,k0+3}
        v2f a;
        a.x = xs[mLocal][k];
        a.y = xs[mLocal][k + 1];
        // B 4x16 fp32 layout mirrors C/D striping: VGPR0 rows {0,2}, VGPR1 rows {1,3}
        v2f b;
        b.x = wsm[k][nCol];
        b.y = wsm[k + 1][nCol];
        c = __builtin_amdgcn_wmma_f32_16x16x4_f32(
            /*neg_a=*/false, a, /*neg_b=*/false, b,
            /*c_mod=*/(short)0, c, /*reuse_a=*/false, /*reuse_b=*/false);
    }

    // C/D layout: VGPR r holds M = half*8 + r, N = lane%16 (per 16x16 f32 table)
    if (nCol < OUT_F) {
        const int mRow = mBase + mSub * 16 + half * 8;
#pragma unroll
        for (int r = 0; r < 8; ++r) {
            h[(mRow + r) * OUT_F + nCol] = c[r];
        }
    }
}

// ---------------------------------------------------------------------------
// Kernel 3: edge scatter-add with HW f32 atomics (L2-resident, device scope).
// 25 threads per edge; each thread moves a float2 of features.
// ---------------------------------------------------------------------------
__device__ __forceinline__ void atom_add_f32(float* p, float v) {
    // CDNA5 GLOBAL_ATOMIC_ADD_F32, no return (STOREcnt), device scope so the
    // RMW is coherent across all WGPs at the shared L2.
    asm volatile("global_atomic_add_f32 %0, %1, off scope:SCOPE_DEV"
                 :: "v"(p), "v"(v)
                 : "memory");
}

__global__ void __launch_bounds__(256)
gcn_scatter_kernel(const float* __restrict__ h,
                   const long long* __restrict__ edge_index,
                   float* __restrict__ out) {
    const unsigned gid = blockIdx.x * blockDim.x + threadIdx.x;
    const unsigned TOT = (unsigned)N_EDGES * 25u;  // 40M threads
    if (gid >= TOT) return;

    const unsigned e = gid / 25u;
    const unsigned j = gid % 25u;
    const int f0 = (int)(2u * j);

    const int src = (int)edge_index[e];
    const int dst = (int)edge_index[(unsigned)N_EDGES + e];

    // 8-byte aligned: row stride 50 floats = 200B, f0 even.
    const float2 v = *(const float2*)(h + (long long)src * OUT_F + f0);
    float* p = out + (long long)dst * OUT_F + f0;
    atom_add_f32(p + 0, v.x);
    atom_add_f32(p + 1, v.y);
}

// ---------------------------------------------------------------------------
// Kernel 4: out = relu(out + b) in place.
// ---------------------------------------------------------------------------
__global__ void gcn_finalize_kernel(float* __restrict__ out,
                                    const float* __restrict__ b) {
    const int i = blockIdx.x * blockDim.x + threadIdx.x;
    const int n = N_NODES * OUT_F;
    if (i < n) {
        float v = out[i] + b[i % OUT_F];
        out[i] = fmaxf(v, 0.f);
    }
}

// ---------------------------------------------------------------------------
extern "C" void kernel_launch(void* const* d_in, const int* in_sizes, int n_in,
                              void* d_out, int out_size, void* d_ws, size_t ws_size,
                              hipStream_t stream) {
    const float* x = (const float*)d_in[0];          // [N_NODES, IN_F]
    const float* W = (const float*)d_in[1];          // [IN_F, OUT_F]
    const float* b = (const float*)d_in[2];          // [OUT_F]
    const long long* edge_index = (const long long*)d_in[3];  // [2, N_EDGES] int64

    float* out = (float*)d_out;                      // [N_NODES, OUT_F]
    float* h = (float*)d_ws;                         // [N_NODES, OUT_F] scratch (20 MB)

    // 1) zero accumulators
    {
        int n4 = (N_NODES * OUT_F) / 4;  // 5,000,000 floats -> 1,250,000 float4
        int blocks = (n4 + 255) / 256;
        gcn_zero_kernel<<<blocks, 256, 0, stream>>>((float4*)out, n4);
    }
    // 2) h = x @ W (WMMA)
    {
        int blocks = N_NODES / 32;  // 3125, exact
        gcn_gemm_kernel<<<blocks, 256, 0, stream>>>(x, W, h);
    }
    // 3) scatter-add over edges
    {
        unsigned total = (unsigned)N_EDGES * 25u;    // 40,000,000
        unsigned blocks = (total + 255u) / 256u;
        gcn_scatter_kernel<<<blocks, 256, 0, stream>>>(h, edge_index, out);
    }
    // 4) bias + relu
    {
        int n = N_NODES * OUT_F;
        int blocks = (n + 255) / 256;
        gcn_finalize_kernel<<<blocks, 256, 0, stream>>>(out, b);
    }
}